// DiceBoundCELoss_11596411699809
// MI455X (gfx1250) — compile-verified
//
#include <hip/hip_runtime.h>
#include <hip/hip_bf16.h>
#include <math.h>

#define B_ 4
#define C_ 8
#define H_ 256
#define W_ 256
#define HW_ (H_*W_)
#define NPIX (B_*H_*W_)      // 262144
#define NPLANE (B_*C_)       // 32
#define TX 16                // columns per vertical tile
#define BIGD 512.0f          // H+W cap, matches reference

typedef __attribute__((ext_vector_type(2))) float v2f;
typedef __attribute__((ext_vector_type(8))) float v8f;

#if defined(__has_builtin)
#  if __has_builtin(__builtin_amdgcn_global_load_async_to_lds_b128)
#    define HAVE_ASYNC_LDS 1
#  endif
#  if __has_builtin(__builtin_amdgcn_s_wait_asynccnt)
#    define HAVE_WAIT_ASYNC 1
#  endif
#endif
#ifndef HAVE_ASYNC_LDS
#  define HAVE_ASYNC_LDS 0
#endif
#ifndef HAVE_WAIT_ASYNC
#  define HAVE_WAIT_ASYNC 0
#endif

// Raw v_sqrt_f32 on the device pass; plain sqrtf keeps the host pass happy
// (kernel bodies are semantically checked in both compilations).
__device__ __forceinline__ float fast_sqrtf(float v) {
#if defined(__AMDGCN__) && defined(__has_builtin)
#  if __has_builtin(__builtin_amdgcn_sqrtf)
    return __builtin_amdgcn_sqrtf(v);
#  else
    return sqrtf(v);
#  endif
#else
    return sqrtf(v);
#endif
}

// Exact pointee type per hipcc diagnostic: GCC-style int 4-vector,
// addrspace(1) global source, addrspace(3) LDS destination.
typedef int v4i_vs __attribute__((vector_size(4 * sizeof(int))));
typedef __attribute__((address_space(1))) v4i_vs* glb_v4i_p;
typedef __attribute__((address_space(3))) v4i_vs* lds_v4i_p;

// accum layout: [0]=ce_sum [1]=intersection [2]=cardinality [3]=bound_sum [4]=valid_b count
__global__ void zero_accum_kernel(float* accum, int* flags) {
    int i = threadIdx.x;
    if (i < 8) accum[i] = 0.0f;
    if (i < NPLANE) flags[i] = 0;
}

// One thread per pixel: log-softmax over C=8, CE / Dice partials, write probs.
__global__ void softmax_ce_kernel(const float* __restrict__ x,
                                  const int*  __restrict__ tgt,
                                  float* __restrict__ probs,
                                  float* accum) {
    __shared__ float red[256][4];
    int p = blockIdx.x * blockDim.x + threadIdx.x;   // exact grid: NPIX threads
    int b = p / HW_;
    int r = p % HW_;
    const float* xp = x + (size_t)b * C_ * HW_ + r;
    float v[C_];
    float m = -INFINITY;
    #pragma unroll
    for (int c = 0; c < C_; ++c) { v[c] = xp[(size_t)c * HW_]; m = fmaxf(m, v[c]); }
    int t = tgt[p];
    int tc = min(max(t, 0), C_ - 1);
    float xt = v[tc];
    float s = 0.0f;
    #pragma unroll
    for (int c = 0; c < C_; ++c) { v[c] = __expf(v[c] - m); s += v[c]; }
    float inv = 1.0f / s;
    float lse = m + __logf(s);
    float* pp = probs + (size_t)b * C_ * HW_ + r;
    float sumP = 0.0f;
    #pragma unroll
    for (int c = 0; c < C_; ++c) { v[c] *= inv; sumP += v[c]; pp[(size_t)c * HW_] = v[c]; }

    float ce_add  = lse - xt;                         // -log p[target]
    float vm      = (t >= 0) ? 1.0f : 0.0f;           // dice valid mask
    float vb      = (t >= 0 && t < C_) ? 1.0f : 0.0f; // boundary valid mask
    int   ts      = (t >= 0) ? tc : 0;
    float inter_add = v[ts] * vm;
    float card_add  = vm * (sumP + 1.0f);

    red[threadIdx.x][0] = ce_add;
    red[threadIdx.x][1] = inter_add;
    red[threadIdx.x][2] = card_add;
    red[threadIdx.x][3] = vb;
    __syncthreads();
    if (threadIdx.x == 0) {
        float a0 = 0.f, a1 = 0.f, a2 = 0.f, a3 = 0.f;
        for (int i = 0; i < 256; ++i) {
            a0 += red[i][0]; a1 += red[i][1]; a2 += red[i][2]; a3 += red[i][3];
        }
        atomicAdd(&accum[0], a0);
        atomicAdd(&accum[1], a1);
        atomicAdd(&accum[2], a2);
        atomicAdd(&accum[4], a3);
    }
}

// One thread per (b,c,y) row: horizontal nearest-seed pass for seed=mask and
// seed=~mask simultaneously; writes capped squared distances g2; sets plane flags.
__global__ void edt_rows_kernel(const int* __restrict__ tgt,
                                float* __restrict__ g2out,
                                float* __restrict__ g2in,
                                int* flags) {
    int id = blockIdx.x * blockDim.x + threadIdx.x;  // 0..B*C*H-1 (exact)
    if (id >= B_ * C_ * H_) return;
    int y = id % H_;
    int c = (id / H_) % C_;
    int b = id / (C_ * H_);
    const int* trow = tgt + ((size_t)b * H_ + y) * W_;
    int plane = b * C_ + c;
    size_t base = (size_t)plane * HW_ + (size_t)y * W_;

    // right-to-left: distance to nearest seed at-or-right (store raw distance)
    float nro = 1e9f, nri = 1e9f;
    int anyo = 0, anyi = 0;
    for (int xx = W_ - 1; xx >= 0; --xx) {
        int t = trow[xx];
        bool valid = (t >= 0) && (t < C_);
        bool so = valid && (t == c);
        bool si = !so;
        if (so) { nro = (float)xx; anyo = 1; }
        if (si) { nri = (float)xx; anyi = 1; }
        g2out[base + xx] = nro - (float)xx;
        g2in [base + xx] = nri - (float)xx;
    }
    // left-to-right: combine with nearest-left, cap at BIG=H+W, square
    float nlo = -1e9f, nli = -1e9f;
    for (int xx = 0; xx < W_; ++xx) {
        int t = trow[xx];
        bool valid = (t >= 0) && (t < C_);
        bool so = valid && (t == c);
        bool si = !so;
        if (so) nlo = (float)xx;
        if (si) nli = (float)xx;
        float d_o = fminf(fminf((float)xx - nlo, g2out[base + xx]), BIGD);
        float d_i = fminf(fminf((float)xx - nli, g2in [base + xx]), BIGD);
        g2out[base + xx] = d_o * d_o;
        g2in [base + xx] = d_i * d_i;
    }
    if (anyo) atomicOr(&flags[plane], 1);   // has_pos
    if (anyi) atomicOr(&flags[plane], 2);   // has_neg
}

// One block per (plane, 16-column tile). Tile filled via
// GLOBAL_LOAD_ASYNC_TO_LDS_B128; single flat min-plus loop over yp updating all
// 16 per-thread outputs (registers); branch-free epilogue (raw v_sqrt_f32,
// multiplicative sign/valid masks) feeding 8 unrolled v_wmma_f32_16x16x4_f32
// accumulations (B = all-ones reduction trick: sum of all lanes' C-components
// equals 16x the true total).
__global__ void edt_col_bound_kernel(const float* __restrict__ g2out,
                                     const float* __restrict__ g2in,
                                     const float* __restrict__ probs,
                                     const int*  __restrict__ tgt,
                                     const int*  __restrict__ flags,
                                     float* accum) {
    __shared__ float so[H_ * TX];
    __shared__ float si[H_ * TX];
    __shared__ float red[256];

    int blk   = blockIdx.x;
    int plane = blk / (W_ / TX);
    int tile  = blk % (W_ / TX);
    int tx0   = tile * TX;
    int b     = plane / C_;
    size_t pbase = (size_t)plane * HW_;

#if HAVE_ASYNC_LDS
    // Async DMA tile fill: 1024 16-byte quads per array, 256 lanes x 4 passes.
    #pragma unroll
    for (int pass = 0; pass < 4; ++pass) {
        int i    = pass * 256 + threadIdx.x;   // quad index
        int row  = i >> 2;
        int quad = i & 3;
        const float* gsrcO = g2out + pbase + (size_t)row * W_ + tx0 + quad * 4;
        const float* gsrcI = g2in  + pbase + (size_t)row * W_ + tx0 + quad * 4;
        float* ldsO = &so[row * TX + quad * 4];
        float* ldsI = &si[row * TX + quad * 4];
        __builtin_amdgcn_global_load_async_to_lds_b128(
            (glb_v4i_p)gsrcO, (lds_v4i_p)ldsO, 0, 0);
        __builtin_amdgcn_global_load_async_to_lds_b128(
            (glb_v4i_p)gsrcI, (lds_v4i_p)ldsI, 0, 0);
    }
#if HAVE_WAIT_ASYNC
    __builtin_amdgcn_s_wait_asynccnt(0);
#else
    asm volatile("s_wait_asynccnt 0" ::: "memory");
#endif
    __syncthreads();
#else
    for (int i = threadIdx.x; i < H_ * TX; i += 256) {
        int yy = i / TX, xc0 = i % TX;
        so[i] = g2out[pbase + (size_t)yy * W_ + tx0 + xc0];
        si[i] = g2in [pbase + (size_t)yy * W_ + tx0 + xc0];
    }
    __syncthreads();
#endif

    int fl = flags[plane];
    float fpos  = (fl & 1) ? 1.0f : 0.0f;   // has_pos ? keep : zero
    float hnegf = (fl & 2) ? 1.0f : 0.0f;   // has_neg ? subtract inner EDT : 0

    int xc   = threadIdx.x & (TX - 1);
    int ygrp = threadIdx.x >> 4;      // 0..15
    int x    = tx0 + xc;

    // 16 outputs per thread: y = ygrp + 16*k, k = 0..15 (register accumulators)
    float d2o[16], d2i[16];
    #pragma unroll
    for (int k = 0; k < 16; ++k) { d2o[k] = 3.4e38f; d2i[k] = 3.4e38f; }

    // Single flat min-plus loop: 2 LDS reads per yp serve all 16 outputs.
    float dyB = (float)ygrp;          // = ygrp - yp, carried in float
    #pragma unroll 4
    for (int yp = 0; yp < H_; ++yp) {
        float go = so[yp * TX + xc];
        float gi = si[yp * TX + xc];
        #pragma unroll
        for (int k = 0; k < 16; ++k) {
            float d = dyB + (float)(16 * k);
            float q = d * d;
            d2o[k] = __builtin_fminf(d2o[k], q + go);
            d2i[k] = __builtin_fminf(d2i[k], q + gi);
        }
        dyB -= 1.0f;
    }

    // Branch-free epilogue: signed = (sqrt(out) - hneg*sqrt(in)) * has_pos,
    // then * probs * per-pixel valid; 8 unrolled WMMA accumulations.
    v8f cacc = {0.f, 0.f, 0.f, 0.f, 0.f, 0.f, 0.f, 0.f};
    v2f bones; bones[0] = 1.0f; bones[1] = 1.0f;
    #pragma unroll
    for (int kk = 0; kk < 8; ++kk) {
        int ya = ygrp + 16 * (2 * kk);
        int yb = ygrp + 16 * (2 * kk + 1);
        float sa = (fast_sqrtf(d2o[2*kk])   - hnegf * fast_sqrtf(d2i[2*kk]))   * fpos;
        float sb = (fast_sqrtf(d2o[2*kk+1]) - hnegf * fast_sqrtf(d2i[2*kk+1])) * fpos;
        int ta = tgt[((size_t)b * H_ + ya) * W_ + x];
        int tb = tgt[((size_t)b * H_ + yb) * W_ + x];
        float vba = (ta >= 0 && ta < C_) ? 1.0f : 0.0f;
        float vbb = (tb >= 0 && tb < C_) ? 1.0f : 0.0f;
        float va = probs[pbase + (size_t)ya * W_ + x] * sa * vba;
        float vb = probs[pbase + (size_t)yb * W_ + x] * sb * vbb;
        v2f a; a[0] = va; a[1] = vb;
        // D = A x ones + C : accumulates the wave-wide sum of all A entries
        cacc = __builtin_amdgcn_wmma_f32_16x16x4_f32(
            false, a, false, bones, (short)0, cacc, false, false);
    }

    float s = cacc[0] + cacc[1] + cacc[2] + cacc[3]
            + cacc[4] + cacc[5] + cacc[6] + cacc[7];
    red[threadIdx.x] = s;
    __syncthreads();
    for (int off = 128; off > 0; off >>= 1) {
        if (threadIdx.x < off) red[threadIdx.x] += red[threadIdx.x + off];
        __syncthreads();
    }
    if (threadIdx.x == 0) {
        atomicAdd(&accum[3], red[0] * (1.0f / 16.0f));  // 16 identical D columns
    }
}

__global__ void finalize_kernel(const float* accum, float* out) {
    if (threadIdx.x == 0 && blockIdx.x == 0) {
        float ce         = accum[0] / (float)NPIX;
        float dice       = 1.0f - (2.0f * accum[1] + 1e-6f) / (accum[2] + 1e-6f);
        float dice_total = 0.1f * ce + 0.9f * dice;
        float bound      = accum[3] / (accum[4] + 1e-8f);
        out[0] = 0.1f * ce + 0.8f * dice_total + 0.1f * bound;
    }
}

extern "C" void kernel_launch(void* const* d_in, const int* in_sizes, int n_in,
                              void* d_out, int out_size, void* d_ws, size_t ws_size,
                              hipStream_t stream) {
    (void)in_sizes; (void)n_in; (void)out_size; (void)ws_size;
    const float* x   = (const float*)d_in[0];   // [4,8,256,256] f32
    const int*   tgt = (const int*)d_in[1];     // [4,256,256] i32
    float* out = (float*)d_out;

    char* ws = (char*)d_ws;
    float* probs = (float*)(ws);                       //  8 MB
    float* g2out = (float*)(ws + (size_t)8388608);     //  8 MB
    float* g2in  = (float*)(ws + (size_t)16777216);    //  8 MB
    float* accum = (float*)(ws + (size_t)25165824);    //  8 floats
    int*   flags = (int*)  (ws + (size_t)25165824 + 256);

    zero_accum_kernel<<<1, 256, 0, stream>>>(accum, flags);
    softmax_ce_kernel<<<NPIX / 256, 256, 0, stream>>>(x, tgt, probs, accum);
    edt_rows_kernel<<<(B_ * C_ * H_) / 256, 256, 0, stream>>>(tgt, g2out, g2in, flags);
    edt_col_bound_kernel<<<NPLANE * (W_ / TX), 256, 0, stream>>>(
        g2out, g2in, probs, tgt, flags, accum);
    finalize_kernel<<<1, 1, 0, stream>>>(accum, out);
}